// INGRAM_88055419503366
// MI455X (gfx1250) — compile-verified
//
#include <hip/hip_runtime.h>
#include <math.h>

typedef __attribute__((ext_vector_type(2))) float v2f;
typedef __attribute__((ext_vector_type(8))) float v8f;

#define D 128
#define NHEAD 8
#define SLOPE 0.2f

__device__ __forceinline__ float leaky(float x) { return x >= 0.f ? x : SLOPE * x; }

__device__ __forceinline__ void atomicMaxF(float* addr, float val) {
  unsigned int* ai = reinterpret_cast<unsigned int*>(addr);
  unsigned int old = __float_as_uint(*addr);
  while (__uint_as_float(old) < val) {
    unsigned int assumed = old;
    old = atomicCAS(ai, assumed, __float_as_uint(val));
    if (old == assumed) break;
  }
}

// ---------------- edge histogram: deg + per-node relation counts ----------------
__global__ void k_hist(const int* __restrict__ ei, const int* __restrict__ et,
                       float* __restrict__ deg, float* __restrict__ cnt,
                       int E_, int nRel) {
  int e = blockIdx.x * blockDim.x + threadIdx.x;
  if (e >= E_) return;
  int dst = ei[E_ + e];
  atomicAdd(&deg[dst], 1.0f);
  atomicAdd(&cnt[(size_t)dst * nRel + et[e]], 1.0f);
}

// ---------------- f32 WMMA GEMM: C(MxN) = A(MxK) @ B(KxN), N==128 ----------------
// one wave computes a 16x64 tile via v_wmma_f32_16x16x4_f32; optional epilogue
// scale 1/max(deg,1) per row.
__global__ void k_gemm_wmma(const float* __restrict__ A, int lda,
                            const float* __restrict__ B, int ldb,
                            float* __restrict__ C, int ldc,
                            int M, int K,
                            const float* __restrict__ degScale) {
  const int lane = threadIdx.x & 31;
  const int wave = (blockIdx.x * blockDim.x + threadIdx.x) >> 5;
  const int tilesN = D / 64;               // N == 128 -> 2
  const int tilesM = (M + 15) >> 4;
  if (wave >= tilesM * tilesN) return;     // wave-uniform: EXEC stays all-1s
  const int tm = wave / tilesN, tn = wave % tilesN;
  const int m0 = tm << 4, n0 = tn << 6;
  const int hf = lane >> 4, l16 = lane & 15;
  int arow = m0 + l16; if (arow > M - 1) arow = M - 1;   // clamp tail rows
  const float* Arow = A + (size_t)arow * lda;

  v8f c0 = {}; v8f c1 = {}; v8f c2 = {}; v8f c3 = {};
  for (int k = 0; k < K; k += 4) {
    const int ka = k + 2 * hf;             // K-split across lane halves per ISA layout
    v2f a; a.x = Arow[ka]; a.y = Arow[ka + 1];
    const float* b0p = B + (size_t)ka * ldb + n0 + l16;
    const float* b1p = b0p + ldb;
    v2f b0, b1, b2, b3;
    b0.x = b0p[0];  b0.y = b1p[0];
    b1.x = b0p[16]; b1.y = b1p[16];
    b2.x = b0p[32]; b2.y = b1p[32];
    b3.x = b0p[48]; b3.y = b1p[48];
    c0 = __builtin_amdgcn_wmma_f32_16x16x4_f32(false, a, false, b0, (short)0, c0, false, false);
    c1 = __builtin_amdgcn_wmma_f32_16x16x4_f32(false, a, false, b1, (short)0, c1, false, false);
    c2 = __builtin_amdgcn_wmma_f32_16x16x4_f32(false, a, false, b2, (short)0, c2, false, false);
    c3 = __builtin_amdgcn_wmma_f32_16x16x4_f32(false, a, false, b3, (short)0, c3, false, false);
  }
#pragma unroll
  for (int i = 0; i < 8; ++i) {
    int row = m0 + i + 8 * hf;
    if (row < M) {
      float s = 1.0f;
      if (degScale) s = 1.0f / fmaxf(degScale[row], 1.0f);
      float* crow = C + (size_t)row * ldc + n0 + l16;
      crow[0]  = c0[i] * s;
      crow[16] = c1[i] * s;
      crow[32] = c2[i] * s;
      crow[48] = c3[i] * s;
    }
  }
}

// ---------------- self attention logits: attn_s = leaky(hP1+hP2+zbP3) @ y_w ------
__global__ void k_self_attn(const float* __restrict__ hP1, const float* __restrict__ hP2,
                            const float* __restrict__ zbP3, const float* __restrict__ yw,
                            float* __restrict__ attnS, float* __restrict__ m, int n) {
  int wave = (blockIdx.x * blockDim.x + threadIdx.x) >> 5;
  int lane = threadIdx.x & 31;
  if (wave >= n) return;
  float4 a1 = ((const float4*)(hP1  + (size_t)wave * D))[lane];
  float4 a2 = ((const float4*)(hP2  + (size_t)wave * D))[lane];
  float4 a3 = ((const float4*)(zbP3 + (size_t)wave * D))[lane];
  float pre[4] = { leaky(a1.x + a2.x + a3.x), leaky(a1.y + a2.y + a3.y),
                   leaky(a1.z + a2.z + a3.z), leaky(a1.w + a2.w + a3.w) };
  float acc[NHEAD];
#pragma unroll
  for (int h = 0; h < NHEAD; ++h) acc[h] = 0.f;
#pragma unroll
  for (int j = 0; j < 4; ++j) {
    const float4* ywr = (const float4*)(yw + (size_t)(4 * lane + j) * NHEAD);
    float4 ylo = ywr[0], yhi = ywr[1];
    acc[0] += pre[j] * ylo.x; acc[1] += pre[j] * ylo.y; acc[2] += pre[j] * ylo.z; acc[3] += pre[j] * ylo.w;
    acc[4] += pre[j] * yhi.x; acc[5] += pre[j] * yhi.y; acc[6] += pre[j] * yhi.z; acc[7] += pre[j] * yhi.w;
  }
#pragma unroll
  for (int off = 16; off; off >>= 1)
#pragma unroll
    for (int h = 0; h < NHEAD; ++h) acc[h] += __shfl_xor(acc[h], off, 32);
  if (lane < NHEAD) {
    attnS[(size_t)wave * NHEAD + lane] = acc[lane];
    m[(size_t)wave * NHEAD + lane] = acc[lane];   // init running max with attn_s
  }
}

// ---------------- edge attention logits + segment max ----------------------------
__global__ void k_edge_attn(const int* __restrict__ ei, const int* __restrict__ et,
                            const float* __restrict__ hP1, const float* __restrict__ hP2,
                            const float* __restrict__ zP3, const float* __restrict__ yw,
                            float* __restrict__ attnE, float* __restrict__ m, int E_) {
  int wave = (blockIdx.x * blockDim.x + threadIdx.x) >> 5;
  int lane = threadIdx.x & 31;
  if (wave >= E_) return;
  int src = ei[wave], dst = ei[E_ + wave], t = et[wave];
  float4 a1 = ((const float4*)(hP1 + (size_t)dst * D))[lane];
  float4 a2 = ((const float4*)(hP2 + (size_t)src * D))[lane];
  float4 a3 = ((const float4*)(zP3 + (size_t)t   * D))[lane];
  float pre[4] = { leaky(a1.x + a2.x + a3.x), leaky(a1.y + a2.y + a3.y),
                   leaky(a1.z + a2.z + a3.z), leaky(a1.w + a2.w + a3.w) };
  float acc[NHEAD];
#pragma unroll
  for (int h = 0; h < NHEAD; ++h) acc[h] = 0.f;
#pragma unroll
  for (int j = 0; j < 4; ++j) {
    const float4* ywr = (const float4*)(yw + (size_t)(4 * lane + j) * NHEAD);
    float4 ylo = ywr[0], yhi = ywr[1];
    acc[0] += pre[j] * ylo.x; acc[1] += pre[j] * ylo.y; acc[2] += pre[j] * ylo.z; acc[3] += pre[j] * ylo.w;
    acc[4] += pre[j] * yhi.x; acc[5] += pre[j] * yhi.y; acc[6] += pre[j] * yhi.z; acc[7] += pre[j] * yhi.w;
  }
#pragma unroll
  for (int off = 16; off; off >>= 1)
#pragma unroll
    for (int h = 0; h < NHEAD; ++h) acc[h] += __shfl_xor(acc[h], off, 32);
  if (lane < NHEAD) {
    float v = acc[lane];
    attnE[(size_t)wave * NHEAD + lane] = v;
    atomicMaxF(&m[(size_t)dst * NHEAD + lane], v);
  }
}

// ---------------- softmax denominator: edge contributions ------------------------
__global__ void k_denom_edge(const int* __restrict__ ei, const float* __restrict__ attnE,
                             const float* __restrict__ m, float* __restrict__ denom, int E_) {
  int idx = blockIdx.x * blockDim.x + threadIdx.x;
  if (idx >= E_ * NHEAD) return;
  int e = idx >> 3, h = idx & 7;
  int dst = ei[E_ + e];
  atomicAdd(&denom[(size_t)dst * NHEAD + h], expf(attnE[idx] - m[(size_t)dst * NHEAD + h]));
}

// ---------------- softmax denominator: self term; stash e_s over attn_s ----------
__global__ void k_self_denom(float* __restrict__ attnS, const float* __restrict__ m,
                             float* __restrict__ denom, int n) {
  int idx = blockIdx.x * blockDim.x + threadIdx.x;
  if (idx >= n * NHEAD) return;
  float es = expf(attnS[idx] - m[idx]);
  attnS[idx] = es;          // reuse buffer: now holds e_s
  denom[idx] += es;         // exclusive per element, no atomic needed
}

// ---------------- weighted edge aggregation into d_out ---------------------------
__global__ void k_edge_agg(const int* __restrict__ ei, const int* __restrict__ et,
                           const float* __restrict__ hWc, const float* __restrict__ zWc,
                           const float* __restrict__ attnE, const float* __restrict__ m,
                           const float* __restrict__ denom, float* __restrict__ out, int E_) {
  int wave = (blockIdx.x * blockDim.x + threadIdx.x) >> 5;
  int lane = threadIdx.x & 31;
  if (wave >= E_) return;
  int src = ei[wave], dst = ei[E_ + wave], t = et[wave];
  int h = lane >> 2;  // 4 cols per lane -> 16 cols per head -> head = lane/4
  size_t dh = (size_t)dst * NHEAD + h;
  float w = expf(attnE[(size_t)wave * NHEAD + h] - m[dh]) / denom[dh];
  float4 tw = ((const float4*)(hWc + (size_t)src * D))[lane];
  float4 tz = ((const float4*)(zWc + (size_t)t   * D))[lane];
  float* o = out + (size_t)dst * D + lane * 4;
  atomicAdd(o + 0, w * (tw.x + tz.x));
  atomicAdd(o + 1, w * (tw.y + tz.y));
  atomicAdd(o + 2, w * (tw.z + tz.z));
  atomicAdd(o + 3, w * (tw.w + tz.w));
}

// ---------------- finalize: agg + w_s*t_s + residual, leaky, deg mask ------------
__global__ void k_final(const float* __restrict__ h, const float* __restrict__ hWc,
                        const float* __restrict__ zbWc, const float* __restrict__ es,
                        const float* __restrict__ denom, const float* __restrict__ deg,
                        const float* __restrict__ rw, float* __restrict__ out, int n) {
  int wave = (blockIdx.x * blockDim.x + threadIdx.x) >> 5;
  int lane = threadIdx.x & 31;
  if (wave >= n) return;
  int hh = lane >> 2;
  size_t ih = (size_t)wave * NHEAD + hh;
  float ws = es[ih] / denom[ih];
  float4 a  = ((const float4*)(out  + (size_t)wave * D))[lane];
  float4 w4 = ((const float4*)(hWc  + (size_t)wave * D))[lane];
  float4 z4 = ((const float4*)(zbWc + (size_t)wave * D))[lane];
  float4 hv = ((const float4*)(h    + (size_t)wave * D))[lane];
  float r = rw[0];
  bool keep = deg[wave] > 0.f;
  float4 o;
  o.x = keep ? leaky(a.x + ws * (w4.x + z4.x) + r * hv.x) : hv.x;
  o.y = keep ? leaky(a.y + ws * (w4.y + z4.y) + r * hv.y) : hv.y;
  o.z = keep ? leaky(a.z + ws * (w4.z + z4.z) + r * hv.z) : hv.z;
  o.w = keep ? leaky(a.w + ws * (w4.w + z4.w) + r * hv.w) : hv.w;
  ((float4*)(out + (size_t)wave * D))[lane] = o;
}

extern "C" void kernel_launch(void* const* d_in, const int* in_sizes, int n_in,
                              void* d_out, int out_size, void* d_ws, size_t ws_size,
                              hipStream_t stream) {
  const float* h  = (const float*)d_in[0];
  const float* z  = (const float*)d_in[1];
  const float* Wc = (const float*)d_in[2];
  const float* P  = (const float*)d_in[3];
  const float* yw = (const float*)d_in[4];
  const float* rw = (const float*)d_in[5];
  const int*   ei = (const int*)d_in[6];
  const int*   et = (const int*)d_in[7];
  const int nEnt = in_sizes[0] / D;
  const int nRel = in_sizes[1] / D;
  const int E_   = in_sizes[6] / 2;
  float* out = (float*)d_out;

  float* ws = (float*)d_ws;
  size_t off = 0;
  auto alloc = [&](size_t nfloats) {
    float* p = ws + off;
    off += (nfloats + 63) & ~(size_t)63;
    return p;
  };
  float* cnt   = alloc((size_t)nEnt * nRel);   // relation histogram per node
  float* deg   = alloc((size_t)nEnt);
  float* hP1   = alloc((size_t)nEnt * D);
  float* hP2   = alloc((size_t)nEnt * D);
  float* hWc   = alloc((size_t)nEnt * D);
  float* zbP3  = alloc((size_t)nEnt * D);      // z_bar @ P3
  float* zbWc  = alloc((size_t)nEnt * D);      // z_bar @ Wc_z
  float* zP3   = alloc((size_t)nRel * D);
  float* zWcb  = alloc((size_t)nRel * D);
  float* attnS = alloc((size_t)nEnt * NHEAD);
  float* mbuf  = alloc((size_t)nEnt * NHEAD);
  float* denom = alloc((size_t)nEnt * NHEAD);
  float* attnE = alloc((size_t)E_ * NHEAD);

  hipMemsetAsync(cnt, 0, (size_t)nEnt * nRel * sizeof(float), stream);
  hipMemsetAsync(deg, 0, (size_t)nEnt * sizeof(float), stream);
  hipMemsetAsync(denom, 0, (size_t)nEnt * NHEAD * sizeof(float), stream);
  hipMemsetAsync(out, 0, (size_t)nEnt * D * sizeof(float), stream);

  k_hist<<<(E_ + 255) / 256, 256, 0, stream>>>(ei, et, deg, cnt, E_, nRel);

  auto gemm = [&](const float* A, int lda, const float* B, float* C,
                  int M, int K, const float* scale) {
    int tiles = ((M + 15) / 16) * (D / 64);
    int blocks = (tiles * 32 + 255) / 256;
    k_gemm_wmma<<<blocks, 256, 0, stream>>>(A, lda, B, D, C, D, M, K, scale);
  };
  gemm(h, D, P,             hP1,  nEnt, D, nullptr);       // h @ P1
  gemm(h, D, P + D * D,     hP2,  nEnt, D, nullptr);       // h @ P2
  gemm(h, D, Wc,            hWc,  nEnt, D, nullptr);       // h @ Wc_h
  gemm(z, D, P + 2 * D * D, zP3,  nRel, D, nullptr);       // z @ P3
  gemm(z, D, Wc + D * D,    zWcb, nRel, D, nullptr);       // z @ Wc_z
  gemm(cnt, nRel, zP3,  zbP3, nEnt, nRel, deg);            // (cnt @ zP3)/max(deg,1)
  gemm(cnt, nRel, zWcb, zbWc, nEnt, nRel, deg);            // (cnt @ zWc)/max(deg,1)

  const int wpb = 8;  // wave32 per 256-thread block
  k_self_attn<<<(nEnt + wpb - 1) / wpb, 256, 0, stream>>>(hP1, hP2, zbP3, yw, attnS, mbuf, nEnt);
  k_edge_attn<<<(E_ + wpb - 1) / wpb, 256, 0, stream>>>(ei, et, hP1, hP2, zP3, yw, attnE, mbuf, E_);
  k_denom_edge<<<((size_t)E_ * NHEAD + 255) / 256, 256, 0, stream>>>(ei, attnE, mbuf, denom, E_);
  k_self_denom<<<((size_t)nEnt * NHEAD + 255) / 256, 256, 0, stream>>>(attnS, mbuf, denom, nEnt);
  k_edge_agg<<<(E_ + wpb - 1) / wpb, 256, 0, stream>>>(ei, et, hWc, zWcb, attnE, mbuf, denom, out, E_);
  k_final<<<(nEnt + wpb - 1) / wpb, 256, 0, stream>>>(h, hWc, zbWc, attnS, denom, deg, rw, out, nEnt);
}